// MCPBRNN_GWVariant_Routing_MRRegular_10196252360848
// MI455X (gfx1250) — compile-verified
//
#include <hip/hip_runtime.h>
#include <stdint.h>

// ---------------------------------------------------------------------------
// MCPBRNN scalar-hidden recurrence: exact sequential scan over B*T steps.
// Nonlinear carry => inherently serial; optimized for per-step latency and
// CDNA5 async global->LDS double-buffered prefetch (ASYNCcnt counted waits).
// ---------------------------------------------------------------------------

#define CHUNK 4096          // floats per LDS chunk (16 KB), multiple of T=16
#define TLEN  16
#define LOADS_PER_CHUNK 32  // 4096 floats / (32 lanes * 4 floats per b128)

__global__ void zero_out_kernel(float* __restrict__ out, int n) {
    int i = blockIdx.x * blockDim.x + threadIdx.x;
    if (i < n) out[i] = 0.0f;
}

__device__ __forceinline__ float fast_exp2(float x) { return __builtin_amdgcn_exp2f(x); }
__device__ __forceinline__ float fast_rcp(float x)  { return __builtin_amdgcn_rcpf(x); }

// Each of the 32 lanes issues LOADS_PER_CHUNK async b128 copies: global -> LDS.
// Global address is clamped to stay inside x (last partial chunk); clamped
// lanes fetch valid-but-unused data so the ASYNCcnt issue count stays fixed
// at 32 per chunk (needed for the counted s_wait_asynccnt below).
__device__ __forceinline__ void issue_chunk_async(const float* __restrict__ x,
                                                  long chunk_flat_start,
                                                  long x_last_ofs,   // x_elems - 4
                                                  unsigned lds_buf_byte_base,
                                                  int lane) {
#pragma unroll
    for (int q = 0; q < LOADS_PER_CHUNK; ++q) {
        long fofs = chunk_flat_start + (long)(q * 32 + lane) * 4;
        if (fofs > x_last_ofs) fofs = x_last_ofs;
        unsigned long long gaddr =
            (unsigned long long)(uintptr_t)(x + fofs);
        unsigned lds_dst = lds_buf_byte_base + (unsigned)(q * 32 + lane) * 16u;
        asm volatile("global_load_async_to_lds_b128 %0, %1, off"
                     :
                     : "v"(lds_dst), "v"(gaddr)
                     : "memory");
    }
}

__global__ void __launch_bounds__(32)
mcpbrnn_scan_kernel(const float* __restrict__ x,
                    const float* __restrict__ c_mean,
                    const float* __restrict__ c_std,
                    const float* __restrict__ Ini_C,
                    const float* __restrict__ w_r_yom,
                    const float* __restrict__ w_r_yfm,
                    const float* __restrict__ w_r_yvm,
                    const float* __restrict__ b0_yom_p,
                    const float* __restrict__ w_b1_yom,
                    const float* __restrict__ w_s_yvm,
                    const float* __restrict__ b0_yrm_p,
                    const int*   __restrict__ time_lag_p,
                    float* __restrict__ out,
                    int Bn) {
    __shared__ float buf[2][CHUNK];

    const int  lane  = threadIdx.x;
    const int  tl    = time_lag_p[0];
    const long total = (long)(Bn - tl) * TLEN;   // always a multiple of 16
    if (total <= 0) return;

    const long x_elems    = (long)Bn * TLEN;
    const long x_last_ofs = x_elems - 4;
    const long start      = (long)tl * TLEN;     // flat index of first step
    const int  nchunks    = (int)((total + CHUNK - 1) / CHUNK);

    // LDS byte base of buf: addrspacecast(local->flat) puts the DS byte
    // offset in the low 32 bits, which is what the async-load VDST
    // (wave-relative LDS address) expects.
    const unsigned lds_base      = (unsigned)(uintptr_t)(&buf[0][0]);
    const unsigned lds_buf_bytes = (unsigned)(CHUNK * sizeof(float));

    // ---- time-invariant gate constants (hoisted; exp2-folded) -------------
    const float LOG2E  = 1.4426950408889634f;
    const float mo     = c_mean[0];
    const float inv_so = 1.0f / c_std[0];
    const float e_om   = expf(w_r_yom[0]);
    const float e_fm   = expf(w_r_yfm[0]);
    const float oo1    = e_om / (e_om + e_fm);
    const float wb1    = w_b1_yom[0];
    const float b0     = b0_yom_p[0];
    const float eb     = expf(b0_yrm_p[0]);
    const float ws     = expf(w_s_yvm[0]);
    const float sv     = 1.0f / (1.0f + expf(-w_r_yvm[0]));
    // sigmoid(z) = 1/(1 + 2^(sig_a*c + sig_b)), z = b0 + (c-mo)*inv_so*wb1
    const float sig_a  = -wb1 * inv_so * LOG2E;
    const float sig_b  = (mo * inv_so * wb1 - b0) * LOG2E;
    // tanh(y) = 1 - 2/(1 + 2^(th_a*c + th_b)), y = (c/500 - eb)*ws
    const float th_a   = ws * (2.0f * LOG2E / 500.0f);
    const float th_b   = -eb * ws * (2.0f * LOG2E);
    const float K      = eb * 500.0f;            // eb * scale_mr

    float c = Ini_C[0];

    // ---- software pipeline: prefetch chunk 0 ------------------------------
    issue_chunk_async(x, start, x_last_ofs, lds_base, lane);

    long done = 0;
    for (int k = 0; k < nchunks; ++k) {
        const int  pb   = k & 1;
        const bool more = (k + 1) < nchunks;

        if (more) {
            issue_chunk_async(x, start + (long)(k + 1) * CHUNK, x_last_ofs,
                              lds_base + (unsigned)((k + 1) & 1) * lds_buf_bytes,
                              lane);
            // async loads complete in order: <=32 outstanding means the 32
            // loads of chunk k have all landed in LDS.
            asm volatile("s_wait_asynccnt 32" ::: "memory");
        } else {
            asm volatile("s_wait_asynccnt 0" ::: "memory");
        }

        const long rem = total - done;
        const int  cnt = (int)(rem < CHUNK ? rem : CHUNK);  // multiple of 16

        if (lane == 0) {
            const long base = start + done;      // 16-aligned flat index
            for (int j0 = 0; j0 < cnt; j0 += TLEN) {
                // Per-row body: 16 dependent steps, fully unrolled so the
                // t==15 store step is static (no per-step compare/branch)
                // and the 16 LDS reads can batch into ds_load_b128s.
                float oo_l = 0.f, f_l = 0.f, ov_l = 0.f, mr_l = 0.f, c_old = 0.f;
#pragma unroll
                for (int t = 0; t < TLEN; ++t) {
                    const float u   = buf[pb][j0 + t];
                    const float e1  = fast_exp2(fmaf(sig_a, c, sig_b));
                    const float oo  = oo1 * fast_rcp(1.0f + e1);
                    const float f   = 1.0f - oo;
                    const float e2  = fast_exp2(fmaf(th_a, c, th_b));
                    const float th  = 1.0f - 2.0f * fast_rcp(1.0f + e2);
                    const float ov1 = sv * th;
                    const float ov  = fminf(ov1, f);   // ov1 - relu(ov1 - f)
                    const float mr  = -ov * fabsf(c - K);
                    if (t == TLEN - 1) {               // static: last step
                        oo_l = oo; f_l = f; ov_l = ov; mr_l = mr; c_old = c;
                    }
                    c = fmaf(f, c, u + mr);
                }
                const long row = (base + j0) >> 4;     // row index b
                out[row]            = oo_l * c_old;    // h_n = oo * c_old
                out[(long)Bn + row] = c_old;           // c_n = c_old
                out[2L * Bn + row]  = oo_l;
                out[3L * Bn + row]  = f_l;
                out[4L * Bn + row]  = ov_l;
                out[5L * Bn + row]  = mr_l;
            }
        }
        done += cnt;
    }
}

extern "C" void kernel_launch(void* const* d_in, const int* in_sizes, int n_in,
                              void* d_out, int out_size, void* d_ws, size_t ws_size,
                              hipStream_t stream) {
    const float* x        = (const float*)d_in[0];
    // d_in[1] = y_obs (unused by forward)
    const float* c_mean   = (const float*)d_in[2];
    const float* c_std    = (const float*)d_in[3];
    const float* Ini_C    = (const float*)d_in[4];
    const float* w_r_yom  = (const float*)d_in[5];
    const float* w_r_yfm  = (const float*)d_in[6];
    const float* w_r_yvm  = (const float*)d_in[7];
    const float* b0_yom   = (const float*)d_in[8];
    const float* w_b1_yom = (const float*)d_in[9];
    const float* w_s_yvm  = (const float*)d_in[10];
    const float* b0_yrm   = (const float*)d_in[11];
    // d_in[12] = epoch (unused)
    const int*   time_lag = (const int*)d_in[13];

    float* out = (float*)d_out;
    const int Bn = in_sizes[0] / TLEN;

    // Rows b < time_lag must stay zero; harness poisons d_out, so clear first.
    zero_out_kernel<<<(out_size + 255) / 256, 256, 0, stream>>>(out, out_size);

    // One wave runs the (inherently serial) nonlinear scan.
    mcpbrnn_scan_kernel<<<1, 32, 0, stream>>>(x, c_mean, c_std, Ini_C,
                                              w_r_yom, w_r_yfm, w_r_yvm,
                                              b0_yom, w_b1_yom, w_s_yvm, b0_yrm,
                                              time_lag, out, Bn);
}